// RayMarcherTaichi_77532749627515
// MI455X (gfx1250) — compile-verified
//
#include <hip/hip_runtime.h>
#include <stdint.h>

// ---------------------------------------------------------------------------
// Occupancy-grid ray marcher (stable-dreamfusion style) for MI455X / gfx1250.
//
//   Phase 0: zero d_out (reference scatters into zeros).
//   Phase 1: pack_kernel  : bitfield int32[gs^3/8] (one byte/elem) -> dense
//            uint32 words (gs^3 bits == 256KB for gs=128) in d_ws.
//   Phase 2: march_kernel : stage the packed grid into LDS with CDNA5
//            global_load_async_to_lds_b128 (+ s_wait_asynccnt); each thread
//            marches one ray with all bit tests hitting LDS (320KB/WGP on
//            CDNA5 holds the whole grid). 64-thread blocks spread the 128
//            wave32s across ~64 WGPs => ~1 wave per SIMD32 for the
//            latency-chain-bound march.
//   Phase 3: scan_kernel  : single-block exclusive scan over counts ->
//            starts, rays_a, total (deterministic compaction order).
//   Phase 4: scatter_kernel: write compacted xyzs/dirs/deltas/ts.
// ---------------------------------------------------------------------------

#define SQRT3F 1.7320508075688772f
#define MIP_BOUND 0.5f /* SCALE */

__device__ __forceinline__ float calc_dt(float t, float gs) {
    // clip(t * EXP_STEP_FACTOR, sqrt3/1024, sqrt3*2*SCALE/gs); EXP_STEP_FACTOR==0
    float x  = 0.0f * t;
    float lo = SQRT3F / 1024.0f;
    float hi = SQRT3F * 2.0f * MIP_BOUND / gs;
    return fminf(fmaxf(x, lo), hi);
}

__device__ __forceinline__ int frexp_e(float x) {
    // exponent e with x = m * 2^e, m in [0.5,1); 0 for x<=0
    if (x > 0.0f) { int e; (void)frexpf(x, &e); return e; }
    return 0;
}

__device__ __forceinline__ uint32_t expand_bits(uint32_t v) {
    v = (v * 0x00010001u) & 0xFF0000FFu;
    v = (v * 0x00000101u) & 0x0F00F00Fu;
    v = (v * 0x00000011u) & 0xC30C30C3u;
    v = (v * 0x00000005u) & 0x49249249u;
    return v;
}

__device__ __forceinline__ uint32_t morton3d(uint32_t x, uint32_t y, uint32_t z) {
    return expand_bits(x) | (expand_bits(y) << 1) | (expand_bits(z) << 2);
}

// ---------------------------------------------------------------------------
// Phase 1: pack low bytes of 4 consecutive int32 bitfield elems into 1 word.
// bit (idx & 31) of packed[idx >> 5]  ==  bit (idx & 7) of bitfield[idx >> 3]
// ---------------------------------------------------------------------------
__global__ void __launch_bounds__(256)
pack_kernel(const int* __restrict__ bitfield, uint32_t* __restrict__ packed, int n_words) {
    int w = blockIdx.x * blockDim.x + threadIdx.x;
    if (w >= n_words) return;
    uint32_t b0 = (uint32_t)bitfield[4 * w + 0] & 0xFFu;
    uint32_t b1 = (uint32_t)bitfield[4 * w + 1] & 0xFFu;
    uint32_t b2 = (uint32_t)bitfield[4 * w + 2] & 0xFFu;
    uint32_t b3 = (uint32_t)bitfield[4 * w + 3] & 0xFFu;
    packed[w] = b0 | (b1 << 8) | (b2 << 16) | (b3 << 24);
}

// ---------------------------------------------------------------------------
// Phase 2: march. One thread per ray; packed grid staged in LDS.
// 64-thread blocks: 2 wave32s/block on separate SIMD32s; 1 block/WGP
// (256KB LDS each), so 64 blocks occupy ~64 WGPs with minimal issue
// contention on the dependent march chain.
// ---------------------------------------------------------------------------
#define MARCH_BLOCK 64

__global__ void __launch_bounds__(MARCH_BLOCK)
march_kernel(const float* __restrict__ ro, const float* __restrict__ rd,
             const float* __restrict__ hits, const float* __restrict__ noise,
             const uint32_t* __restrict__ packed,
             const int* __restrict__ casc_p, const int* __restrict__ gs_p,
             float* __restrict__ ts_buf, int* __restrict__ counts,
             int N, int max_samples, int bf_bytes) {
    extern __shared__ uint32_t smem[];

#if defined(__gfx1250__)
    // CDNA5 async global->LDS staging: each active lane moves 16B/instruction,
    // tracked on ASYNCcnt (HW throttles at counter saturation).
    {
        uint32_t lds_base = (uint32_t)(uintptr_t)(void*)smem; // low 32b of flat addr == LDS offset
        for (int ofs = (int)threadIdx.x * 16; ofs < bf_bytes; ofs += (int)blockDim.x * 16) {
            uint32_t laddr      = lds_base + (uint32_t)ofs;
            const void* gaddr   = (const char*)packed + ofs;
            asm volatile("global_load_async_to_lds_b128 %0, %1, off"
                         :: "v"(laddr), "v"(gaddr) : "memory");
        }
        asm volatile("s_wait_asynccnt 0" ::: "memory");
    }
#else
    for (int i = (int)threadIdx.x; i < bf_bytes / 4; i += (int)blockDim.x)
        smem[i] = packed[i];
#endif
    __syncthreads();

    int ray = blockIdx.x * blockDim.x + threadIdx.x;
    if (ray >= N) return;

    const int   casc = *casc_p;
    const int   gsi  = *gs_p;
    const float gs   = (float)gsi;
    const uint32_t gs3 = (uint32_t)(gsi * gsi * gsi);

    float ox = ro[3 * ray + 0], oy = ro[3 * ray + 1], oz = ro[3 * ray + 2];
    float dx = rd[3 * ray + 0], dy = rd[3 * ray + 1], dz = rd[3 * ray + 2];
    float t1 = hits[2 * ray + 0], t2 = hits[2 * ray + 1];
    float nzr = noise[ray];

    float dix = 1.0f / dx, diy = 1.0f / dy, diz = 1.0f / dz;

    float t = (t1 >= 0.0f) ? (t1 + calc_dt(t1, gs) * nzr) : t1;
    int   n = 0;

    while (t >= 0.0f && t < t2 && n < max_samples) {
        float x  = ox + t * dx;
        float y  = oy + t * dy;
        float z  = oz + t * dz;
        float dt = calc_dt(t, gs);

        float amax = fmaxf(fabsf(x), fmaxf(fabsf(y), fabsf(z)));
        int mip_p = min(max(frexp_e(amax) + 1, 0), casc - 1);
        int mip_d = min(max(frexp_e(dt * gs), 0), casc - 1);
        int mip   = max(mip_p, mip_d);

        float nx = fminf(fmaxf(0.5f * (x / MIP_BOUND + 1.0f) * gs, 0.0f), gs - 1.0f);
        float ny = fminf(fmaxf(0.5f * (y / MIP_BOUND + 1.0f) * gs, 0.0f), gs - 1.0f);
        float nz = fminf(fmaxf(0.5f * (z / MIP_BOUND + 1.0f) * gs, 0.0f), gs - 1.0f);

        uint32_t idx  = (uint32_t)mip * gs3 +
                        morton3d((uint32_t)nx, (uint32_t)ny, (uint32_t)nz);
        uint32_t word = smem[idx >> 5];          // LDS-resident bit test
        bool occ = ((word >> (idx & 31u)) & 1u) != 0u;

        if (occ) {
            ts_buf[(size_t)ray * (size_t)max_samples + n] = t;
            n += 1;
            t = t + dt;
        } else {
            // empty-space skip to next cell boundary, in dt-sized steps
            float sx = (dx > 0.0f) ? 1.0f : ((dx < 0.0f) ? -1.0f : 0.0f);
            float sy = (dy > 0.0f) ? 1.0f : ((dy < 0.0f) ? -1.0f : 0.0f);
            float sz = (dz > 0.0f) ? 1.0f : ((dz < 0.0f) ? -1.0f : 0.0f);
            float tx = (((nx + 0.5f + 0.5f * sx) / gs * 2.0f - 1.0f) * MIP_BOUND - x) * dix;
            float ty = (((ny + 0.5f + 0.5f * sy) / gs * 2.0f - 1.0f) * MIP_BOUND - y) * diy;
            float tz = (((nz + 0.5f + 0.5f * sz) / gs * 2.0f - 1.0f) * MIP_BOUND - z) * diz;
            float t_target = t + fmaxf(0.0f, fminf(tx, fminf(ty, tz)));
            float tt = t + calc_dt(t, gs);
            while (tt < t_target) tt += calc_dt(tt, gs);
            t = tt;
        }
    }
    counts[ray] = n;
}

// ---------------------------------------------------------------------------
// Phase 3: single-block exclusive scan over per-ray counts.
// Writes starts[], rays_a (int32 triplets at head of d_out) and total.
// ---------------------------------------------------------------------------
__global__ void __launch_bounds__(1024)
scan_kernel(const int* __restrict__ counts, int* __restrict__ starts,
            int* __restrict__ out_i, int N, int total_slot) {
    __shared__ int sh[1024];
    const int tid = (int)threadIdx.x;
    const int T = 1024;
    const int items = (N + T - 1) / T;
    const int base = tid * items;

    int sum = 0;
    for (int k = 0; k < items; ++k) {
        int i = base + k;
        if (i < N) sum += counts[i];
    }
    sh[tid] = sum;
    __syncthreads();
    for (int off = 1; off < T; off <<= 1) {
        int v = (tid >= off) ? sh[tid - off] : 0;
        __syncthreads();
        sh[tid] += v;
        __syncthreads();
    }
    int run = sh[tid] - sum; // exclusive prefix for this thread's chunk
    for (int k = 0; k < items; ++k) {
        int i = base + k;
        if (i < N) {
            int c = counts[i];
            starts[i]        = run;
            out_i[3 * i + 0] = i;
            out_i[3 * i + 1] = run;
            out_i[3 * i + 2] = c;
            run += c;
        }
    }
    if (tid == T - 1) out_i[total_slot] = sh[T - 1];
}

// ---------------------------------------------------------------------------
// Phase 4: scatter compacted samples.
// ---------------------------------------------------------------------------
__global__ void __launch_bounds__(256)
scatter_kernel(const float* __restrict__ ro, const float* __restrict__ rd,
               const float* __restrict__ ts_buf,
               const int* __restrict__ counts, const int* __restrict__ starts,
               const int* __restrict__ gs_p,
               float* __restrict__ xyzs, float* __restrict__ dirs,
               float* __restrict__ deltas, float* __restrict__ ts,
               int max_samples) {
    const int ray = (int)blockIdx.x;
    const int ns  = counts[ray];
    const int st  = starts[ray];
    const float gs = (float)(*gs_p);
    const float ox = ro[3 * ray + 0], oy = ro[3 * ray + 1], oz = ro[3 * ray + 2];
    const float dx = rd[3 * ray + 0], dy = rd[3 * ray + 1], dz = rd[3 * ray + 2];

    for (int i = (int)threadIdx.x; i < ns; i += (int)blockDim.x) {
        float t  = ts_buf[(size_t)ray * (size_t)max_samples + i];
        float dt = calc_dt(t, gs);
        int o = st + i;
        xyzs[3 * o + 0] = ox + t * dx;
        xyzs[3 * o + 1] = oy + t * dy;
        xyzs[3 * o + 2] = oz + t * dz;
        dirs[3 * o + 0] = dx;
        dirs[3 * o + 1] = dy;
        dirs[3 * o + 2] = dz;
        deltas[o] = dt;
        ts[o]     = t;
    }
}

// ---------------------------------------------------------------------------
extern "C" void kernel_launch(void* const* d_in, const int* in_sizes, int n_in,
                              void* d_out, int out_size, void* d_ws, size_t ws_size,
                              hipStream_t stream) {
    const float* ro       = (const float*)d_in[0];
    const float* rd       = (const float*)d_in[1];
    const float* hits     = (const float*)d_in[2];
    const int*   bitfield = (const int*)d_in[3];
    const float* noise    = (const float*)d_in[4];
    const int*   casc_p   = (const int*)d_in[5];
    const int*   gs_p     = (const int*)d_in[6];
    // d_in[7] = max_samples (derived host-side from out_size instead)

    const int N = in_sizes[4];                              // = #rays (noise length)
    const long long S = ((long long)out_size - 3LL * N - 1LL) / 8LL; // N*max_samples
    const int max_samples = (int)(S / N);
    const int bf_elems = in_sizes[3];                       // cascades*gs^3/8 int32 elems
    const int n_words  = bf_elems / 4;                      // packed uint32 words
    const int lds_bytes = n_words * 4;                      // 256KB for gs=128 (<=320KB WGP LDS)

    // workspace layout
    uint32_t* packed = (uint32_t*)d_ws;
    int*      counts = (int*)((char*)d_ws + (size_t)n_words * 4u);
    int*      starts = counts + N;
    float*    ts_buf = (float*)(starts + N);                // N*max_samples floats

    // output layout (flat, return order)
    float* out_f  = (float*)d_out;
    int*   out_i  = (int*)d_out;
    float* xyzs   = out_f + 3LL * N;
    float* dirs   = xyzs + 3LL * S;
    float* deltas = dirs + 3LL * S;
    float* ts     = deltas + S;

    hipMemsetAsync(d_out, 0, (size_t)out_size * sizeof(float), stream);

    pack_kernel<<<(n_words + 255) / 256, 256, 0, stream>>>(bitfield, packed, n_words);

    (void)hipFuncSetAttribute((const void*)march_kernel,
                              hipFuncAttributeMaxDynamicSharedMemorySize, lds_bytes);
    march_kernel<<<(N + MARCH_BLOCK - 1) / MARCH_BLOCK, MARCH_BLOCK, lds_bytes, stream>>>(
        ro, rd, hits, noise, packed, casc_p, gs_p, ts_buf, counts,
        N, max_samples, lds_bytes);

    scan_kernel<<<1, 1024, 0, stream>>>(counts, starts, out_i, N, out_size - 1);

    scatter_kernel<<<N, 256, 0, stream>>>(ro, rd, ts_buf, counts, starts, gs_p,
                                          xyzs, dirs, deltas, ts, max_samples);
}